// HSIC_27805618274725
// MI455X (gfx1250) — compile-verified
//
#include <hip/hip_runtime.h>
#include <hip/hip_bf16.h>
#include <math.h>

#define N 4096
#define NNE (16777216u)  // N*N

typedef float v2f __attribute__((ext_vector_type(2)));
typedef float v8f __attribute__((ext_vector_type(8)));

// ---- order-preserving float<->uint key maps (for radix select) ----
__device__ __forceinline__ unsigned f2k(float f) {
  unsigned u = __float_as_uint(f);
  return u ^ ((u & 0x80000000u) ? 0xFFFFFFFFu : 0x80000000u);
}
__device__ __forceinline__ float k2f(unsigned k) {
  unsigned u = (k & 0x80000000u) ? (k ^ 0x80000000u) : ~k;
  return __uint_as_float(u);
}

// ---- squared row norms: one thread per row (64 cols) ----
__global__ void sqnorm_ker(const float* __restrict__ X, float* __restrict__ G) {
  int row = blockIdx.x * 256 + threadIdx.x;
  const float4* p = (const float4*)(X + (size_t)row * 64);
  float s = 0.f;
#pragma unroll
  for (int q = 0; q < 16; ++q) {
    float4 v = p[q];
    s += v.x * v.x + v.y * v.y + v.z * v.z + v.w * v.w;
  }
  G[row] = s;
}

// ---- D = g_i + g_j - 2*X*X^T via fp32 WMMA 16x16x4; one wave per tile ----
__global__ __launch_bounds__(256) void gemm_dist_ker(const float* __restrict__ X,
                                                     const float* __restrict__ G,
                                                     float* __restrict__ D) {
  int wave = threadIdx.x >> 5;
  int lane = threadIdx.x & 31;
  int tile = blockIdx.x * 8 + wave;     // 256*256 tiles
  int tm = tile >> 8;
  int tn = tile & 255;
  int hh  = lane >> 4;                  // half-wave: 0 or 1
  int l16 = lane & 15;

  // A layout (16x4 f32): lanes 0-15 hold K={k0,k0+1}, lanes 16-31 K={k0+2,k0+3}, M=lane%16
  // B = X^T (4x16): identical indexing with N=lane%16 (mirror of A)
  const float* pa = X + ((size_t)(tm * 16 + l16)) * 64 + hh * 2;
  const float* pb = X + ((size_t)(tn * 16 + l16)) * 64 + hh * 2;

  v8f c = {0.f, 0.f, 0.f, 0.f, 0.f, 0.f, 0.f, 0.f};
#pragma unroll
  for (int k0 = 0; k0 < 64; k0 += 4) {
    v2f a = *(const v2f*)(pa + k0);
    v2f b = *(const v2f*)(pb + k0);
    c = __builtin_amdgcn_wmma_f32_16x16x4_f32(false, a, false, b, (short)0, c,
                                              false, false);
  }

  // C/D layout: VGPR r -> M = r + 8*half, N = lane%16
  int col = tn * 16 + l16;
  float gc = G[col];
#pragma unroll
  for (int r = 0; r < 8; ++r) {
    int row = tm * 16 + r + hh * 8;
    D[(size_t)row * N + col] = G[row] + gc - 2.0f * c[r];
  }
}

// ---- radix-select over strict upper triangle ----
__global__ void init_sel_ker(unsigned* selst, unsigned rank) {
  selst[0] = 0u;     // prefix
  selst[1] = rank;   // remaining rank within prefix group
}

__global__ void zero_hist_ker(unsigned* hist) { hist[threadIdx.x] = 0u; }

__global__ __launch_bounds__(256) void histo_ker(const float* __restrict__ D,
                                                 const unsigned* __restrict__ selst,
                                                 unsigned* __restrict__ hist,
                                                 int shift, unsigned mask) {
  __shared__ unsigned lh[256];
  lh[threadIdx.x] = 0u;
  __syncthreads();
  unsigned prefix = selst[0];
  unsigned stride = gridDim.x * 256u;
  for (unsigned idx = blockIdx.x * 256u + threadIdx.x; idx < NNE; idx += stride) {
    unsigned i = idx >> 12, j = idx & 4095u;
    if (j > i) {
      unsigned key = f2k(D[idx]);
      if ((key & mask) == prefix) atomicAdd(&lh[(key >> shift) & 255u], 1u);
    }
  }
  __syncthreads();
  atomicAdd(&hist[threadIdx.x], lh[threadIdx.x]);
}

__global__ void pick_ker(unsigned* selst, const unsigned* hist, int shift,
                         float* result) {
  unsigned r = selst[1];
  unsigned cum = 0u, b = 0u;
  for (; b < 255u; ++b) {
    unsigned c = hist[b];
    if (cum + c > r) break;
    cum += c;
  }
  selst[0] |= b << shift;
  selst[1] = r - cum;
  if (shift == 0) *result = k2f(selst[0]);
}

// width: median = (v1+v2)/2 ; w^2 = median/2 ; K=exp(-D/(2w^2)) -> inv = 2/(v1+v2)
__global__ void width_ker(float* scal, int which) {
  scal[which] = 2.0f / (scal[2] + scal[3]);
}

// ---- K = exp(-D*inv) in place, float4 vectorized ----
__global__ __launch_bounds__(256) void exp_ker(float* __restrict__ D,
                                               const float* __restrict__ scal,
                                               int which) {
  float inv = scal[which];
  size_t idx = ((size_t)blockIdx.x * 256 + threadIdx.x) * 4;
  float4* p = (float4*)(D + idx);
  float4 v = *p;
  v.x = expf(-v.x * inv);
  v.y = expf(-v.y * inv);
  v.z = expf(-v.z * inv);
  v.w = expf(-v.w * inv);
  *p = v;
}

// ---- deterministic row sums: one block per row ----
__global__ __launch_bounds__(256) void rowsum_ker(const float* __restrict__ K,
                                                  float* __restrict__ rs) {
  const float* p = K + (size_t)blockIdx.x * N;
  float s = 0.f;
  for (int j = threadIdx.x; j < N; j += 256) s += p[j];
  __shared__ float sh[256];
  sh[threadIdx.x] = s;
  __syncthreads();
  for (int off = 128; off > 0; off >>= 1) {
    if (threadIdx.x < off) sh[threadIdx.x] += sh[threadIdx.x + off];
    __syncthreads();
  }
  if (threadIdx.x == 0) rs[blockIdx.x] = sh[0];
}

// ---- total sum (from rowsums) and trace, fixed order ----
__global__ __launch_bounds__(256) void sumdiag_ker(const float* __restrict__ K,
                                                   const float* __restrict__ rs,
                                                   float* __restrict__ outp) {
  __shared__ float shs[256], sht[256];
  float s = 0.f, t = 0.f;
  for (int i = threadIdx.x; i < N; i += 256) {
    s += rs[i];
    t += K[(size_t)i * N + i];
  }
  shs[threadIdx.x] = s;
  sht[threadIdx.x] = t;
  __syncthreads();
  for (int off = 128; off > 0; off >>= 1) {
    if (threadIdx.x < off) {
      shs[threadIdx.x] += shs[threadIdx.x + off];
      sht[threadIdx.x] += sht[threadIdx.x + off];
    }
    __syncthreads();
  }
  if (threadIdx.x == 0) {
    outp[0] = shs[0];  // sum
    outp[1] = sht[0];  // trace
  }
}

// ---- fused centered reduction: p1=sum(Kc*Lc), p2=sum((Kc*Lc/6)^2), p3=diag of V ----
__global__ __launch_bounds__(256) void hsic_reduce_ker(
    const float* __restrict__ Km, const float* __restrict__ Lm,
    const float* __restrict__ rsK, const float* __restrict__ rsL,
    const float* __restrict__ scal, float* __restrict__ part) {
  const float invn = 1.0f / (float)N;
  int i = blockIdx.x;
  float tmK = scal[4] * invn * invn;
  float tmL = scal[6] * invn * invn;
  float rKi = rsK[i] * invn, rLi = rsL[i] * invn;
  const float* kp = Km + (size_t)i * N;
  const float* lp = Lm + (size_t)i * N;
  float p1 = 0.f, p2 = 0.f, p3 = 0.f;
  for (int j = threadIdx.x; j < N; j += 256) {
    float kc = kp[j] - rKi - rsK[j] * invn + tmK;
    float lc = lp[j] - rLi - rsL[j] * invn + tmL;
    float m = kc * lc;
    p1 += m;
    float v = m * (1.0f / 6.0f);
    v = v * v;
    p2 += v;
    if (j == i) p3 = v;
  }
  __shared__ float s1[256], s2[256], s3[256];
  s1[threadIdx.x] = p1;
  s2[threadIdx.x] = p2;
  s3[threadIdx.x] = p3;
  __syncthreads();
  for (int off = 128; off > 0; off >>= 1) {
    if (threadIdx.x < off) {
      s1[threadIdx.x] += s1[threadIdx.x + off];
      s2[threadIdx.x] += s2[threadIdx.x + off];
      s3[threadIdx.x] += s3[threadIdx.x + off];
    }
    __syncthreads();
  }
  if (threadIdx.x == 0) {
    part[i * 4 + 0] = s1[0];
    part[i * 4 + 1] = s2[0];
    part[i * 4 + 2] = s3[0];
  }
}

__global__ __launch_bounds__(256) void part_reduce_ker(const float* __restrict__ part,
                                                       float* __restrict__ red) {
  float p1 = 0.f, p2 = 0.f, p3 = 0.f;
  for (int i = threadIdx.x; i < N; i += 256) {
    p1 += part[i * 4 + 0];
    p2 += part[i * 4 + 1];
    p3 += part[i * 4 + 2];
  }
  __shared__ float s1[256], s2[256], s3[256];
  s1[threadIdx.x] = p1;
  s2[threadIdx.x] = p2;
  s3[threadIdx.x] = p3;
  __syncthreads();
  for (int off = 128; off > 0; off >>= 1) {
    if (threadIdx.x < off) {
      s1[threadIdx.x] += s1[threadIdx.x + off];
      s2[threadIdx.x] += s2[threadIdx.x + off];
      s3[threadIdx.x] += s3[threadIdx.x + off];
    }
    __syncthreads();
  }
  if (threadIdx.x == 0) {
    red[0] = s1[0];
    red[1] = s2[0];
    red[2] = s3[0];
  }
}

// ---- regularized lower incomplete gamma P(a,x), double precision ----
__device__ double gammainc_P(double a, double x) {
  if (x <= 0.0) return 0.0;
  double lg = lgamma(a);
  if (x < a + 1.0) {  // series
    double ap = a, sum = 1.0 / a, del = sum;
    for (int i = 0; i < 500; ++i) {
      ap += 1.0;
      del *= x / ap;
      sum += del;
      if (fabs(del) < fabs(sum) * 1e-16) break;
    }
    return sum * exp(-x + a * log(x) - lg);
  } else {  // Lentz continued fraction for Q
    const double tiny = 1e-300;
    double b = x + 1.0 - a, c = 1.0 / tiny, d = 1.0 / b, h = d;
    for (int i = 1; i <= 500; ++i) {
      double an = -(double)i * ((double)i - a);
      b += 2.0;
      d = an * d + b;
      if (fabs(d) < tiny) d = tiny;
      c = b + an / c;
      if (fabs(c) < tiny) c = tiny;
      d = 1.0 / d;
      double delta = d * c;
      h *= delta;
      if (fabs(delta - 1.0) < 1e-16) break;
    }
    double Q = exp(-x + a * log(x) - lg) * h;
    return 1.0 - Q;
  }
}

__device__ double gamma_ppf(double a, double p) {
  double lo = 0.0, hi = a + 10.0 * sqrt(a) + 100.0;
  for (int it = 0; it < 100; ++it) {  // matches reference's 100-iter bisection
    double mid = 0.5 * (lo + hi);
    if (gammainc_P(a, mid) < p) lo = mid; else hi = mid;
  }
  return 0.5 * (lo + hi);
}

__global__ void final_ker(const float* __restrict__ scal,
                          const float* __restrict__ red, float* __restrict__ out) {
  double n = (double)N;
  double s1 = red[0], s2 = red[1], s3 = red[2];
  double testStat = s1 / n;
  double varHSIC = (s2 - s3) / n / (n - 1.0);
  varHSIC = varHSIC * 72.0 * (n - 4.0) * (n - 5.0) / n / (n - 1.0) / (n - 2.0) / (n - 3.0);
  double K0 = (double)scal[4] - (double)scal[5];
  double L0 = (double)scal[6] - (double)scal[7];
  double muX = K0 / n / (n - 1.0);
  double muY = L0 / n / (n - 1.0);
  double mHSIC = (1.0 + muX * muY - muX - muY) / n;
  double al = mHSIC * mHSIC / varHSIC;
  double bet = varHSIC * n / mHSIC;
  double thresh = bet * gamma_ppf(al, 0.2);  // 1 - ALPH
  out[0] = (float)testStat;
  out[1] = (float)thresh;
}

extern "C" void kernel_launch(void* const* d_in, const int* in_sizes, int n_in,
                              void* d_out, int out_size, void* d_ws, size_t ws_size,
                              hipStream_t stream) {
  const float* X = (const float*)d_in[0];
  const float* Y = (const float*)d_in[1];
  float* out = (float*)d_out;

  float* Dx = (float*)d_ws;            // 16,777,216 floats (64 MB) -> becomes K
  float* Dy = Dx + (size_t)N * N;      // 64 MB -> becomes L
  float* Gx = Dy + (size_t)N * N;      // N
  float* Gy = Gx + N;                  // N
  float* rsK = Gy + N;                 // N
  float* rsL = rsK + N;                // N
  float* scal = rsL + N;               // [0]=invX [1]=invY [2]=v1 [3]=v2
                                       // [4]=sumK [5]=trK [6]=sumL [7]=trL
  float* red = scal + 16;              // [0..2] fused reduction results
  unsigned* hist = (unsigned*)(scal + 64);   // 256
  unsigned* selst = hist + 256;              // 2
  float* part = (float*)(selst + 16);        // N*4

  const unsigned RANK1 = 4193279u;  // n(n-1)/2 = 8,386,560 -> middle pair
  const unsigned RANK2 = 4193280u;

  sqnorm_ker<<<N / 256, 256, 0, stream>>>(X, Gx);
  sqnorm_ker<<<N / 256, 256, 0, stream>>>(Y, Gy);
  gemm_dist_ker<<<(256 * 256) / 8, 256, 0, stream>>>(X, Gx, Dx);
  gemm_dist_ker<<<(256 * 256) / 8, 256, 0, stream>>>(Y, Gy, Dy);

  for (int m = 0; m < 2; ++m) {
    float* D = m ? Dy : Dx;
    for (int r = 0; r < 2; ++r) {
      init_sel_ker<<<1, 1, 0, stream>>>(selst, r ? RANK2 : RANK1);
      for (int shift = 24; shift >= 0; shift -= 8) {
        zero_hist_ker<<<1, 256, 0, stream>>>(hist);
        unsigned mask = (shift == 24) ? 0u : (0xFFFFFFFFu << (shift + 8));
        histo_ker<<<1024, 256, 0, stream>>>(D, selst, hist, shift, mask);
        pick_ker<<<1, 1, 0, stream>>>(selst, hist, shift, scal + 2 + r);
      }
    }
    width_ker<<<1, 1, 0, stream>>>(scal, m);
  }

  exp_ker<<<NNE / 4 / 256, 256, 0, stream>>>(Dx, scal, 0);
  exp_ker<<<NNE / 4 / 256, 256, 0, stream>>>(Dy, scal, 1);

  rowsum_ker<<<N, 256, 0, stream>>>(Dx, rsK);
  rowsum_ker<<<N, 256, 0, stream>>>(Dy, rsL);
  sumdiag_ker<<<1, 256, 0, stream>>>(Dx, rsK, scal + 4);
  sumdiag_ker<<<1, 256, 0, stream>>>(Dy, rsL, scal + 6);

  hsic_reduce_ker<<<N, 256, 0, stream>>>(Dx, Dy, rsK, rsL, scal, part);
  part_reduce_ker<<<1, 256, 0, stream>>>(part, red);
  final_ker<<<1, 1, 0, stream>>>(scal, red, out);
}